// Decoder_16415365005695
// MI455X (gfx1250) — compile-verified
//
#include <hip/hip_runtime.h>

typedef float v2f __attribute__((ext_vector_type(2)));
typedef float v8f __attribute__((ext_vector_type(8)));

#define NN 100000
#define NE 1600000

// ---------------- utility kernels ----------------

__global__ void k_zero(float* __restrict__ p, int n) {
  int i = blockIdx.x * blockDim.x + threadIdx.x;
  int stride = gridDim.x * blockDim.x;
  for (; i < n; i += stride) p[i] = 0.0f;
}

__global__ void k_degree(const long long* __restrict__ col, float* __restrict__ deg, int E) {
  int i = blockIdx.x * blockDim.x + threadIdx.x;
  int stride = gridDim.x * blockDim.x;
  for (; i < E; i += stride) atomicAdd(&deg[(int)col[i]], 1.0f);
}

__global__ void k_dinv(const float* __restrict__ deg, float* __restrict__ dinv, int n) {
  int i = blockIdx.x * blockDim.x + threadIdx.x;
  int stride = gridDim.x * blockDim.x;
  for (; i < n; i += stride) dinv[i] = rsqrtf(deg[i] + 1.0f);
}

// ---------------- fp32 WMMA GEMM: H = X(N x FIN) * W(FIN x FOUT) ----------------
// W staged in LDS transposed + zero padded: Wlds[n][k], n < FOUT_P, k < FIN_P.
// One wave per 16x16 output tile, V_WMMA_F32_16X16X4_F32 over K in steps of 4.
// A 16x4 frag (v2f): lanes 0-15 -> M=lane&15, K=k0+{0,1}; lanes 16-31 -> K=k0+{2,3}.
// B 4x16  frag (v2f): lanes 0-15 -> N=lane&15, K=k0+{0,1}; lanes 16-31 -> K=k0+{2,3}.
// C/D (v8f): vgpr v -> M = v + 8*(lane>=16), N = lane&15.
template <int FIN, int FOUT>
__global__ __launch_bounds__(256) void k_gemm_wmma(const float* __restrict__ X,
                                                   const float* __restrict__ W,
                                                   float* __restrict__ H,
                                                   int totalTiles) {
  constexpr int FIN_P = (FIN + 3) & ~3;
  constexpr int FOUT_P = (FOUT + 15) & ~15;
  constexpr int COL_TILES = FOUT_P / 16;
  __shared__ float Wlds[FOUT_P * FIN_P];

  // cooperative staging: transpose + zero-pad
  for (int idx = threadIdx.x; idx < FOUT_P * FIN_P; idx += blockDim.x) {
    int n = idx / FIN_P;
    int k = idx - n * FIN_P;
    float v = 0.0f;
    if (n < FOUT && k < FIN) v = W[k * FOUT + n];
    Wlds[idx] = v;
  }
  __syncthreads();

  int wave = (int)((blockIdx.x * blockDim.x + threadIdx.x) >> 5);
  if (wave < totalTiles) {
    int lane = threadIdx.x & 31;
    int half = lane >> 4;
    int lrow = lane & 15;
    int m0 = (wave / COL_TILES) * 16;  // NN % 16 == 0
    int n0 = (wave % COL_TILES) * 16;
    int ncol = n0 + lrow;              // < FOUT_P always

    v8f c = {0.f, 0.f, 0.f, 0.f, 0.f, 0.f, 0.f, 0.f};
    const float* xr = X + (size_t)(m0 + lrow) * FIN;
    const float* wr = &Wlds[ncol * FIN_P];

#pragma unroll
    for (int k0 = 0; k0 < FIN_P; k0 += 4) {
      int ka = k0 + 2 * half;
      v2f a;
      if (k0 + 4 <= FIN) {             // compile-time: full step, no masking
        a.x = xr[ka];
        a.y = xr[ka + 1];
      } else {                         // tail: clamped loads + cndmask (no exec masking)
        int i0 = (ka < FIN) ? ka : 0;
        int i1 = (ka + 1 < FIN) ? ka + 1 : 0;
        float t0 = xr[i0];
        float t1 = xr[i1];
        a.x = (ka < FIN) ? t0 : 0.0f;
        a.y = (ka + 1 < FIN) ? t1 : 0.0f;
      }
      v2f b = *(const v2f*)(wr + ka);  // 8B-aligned ds_load (padding makes OOB cols zero)
      c = __builtin_amdgcn_wmma_f32_16x16x4_f32(
          false, a, false, b, (short)0, c, false, false);
    }

    if (FOUT == FOUT_P || ncol < FOUT) {
      float* hb = H + (size_t)(m0 + 8 * half) * FOUT + ncol;
#pragma unroll
      for (int v = 0; v < 8; ++v) hb[(size_t)v * FOUT] = c[v];
    }
  }
}

// ---------------- edge scatter: agg[col] += dinv[row]*dinv[col] * h[row] ----------------
// Each wave owns 32 edges: indices/norms loaded coalesced, broadcast via readlane.
template <int F>
__global__ __launch_bounds__(256) void k_scatter(const long long* __restrict__ rows,
                                                 const long long* __restrict__ cols,
                                                 const float* __restrict__ dinv,
                                                 const float* __restrict__ H,
                                                 float* __restrict__ agg, int E) {
  int wave = (int)((blockIdx.x * blockDim.x + threadIdx.x) >> 5);
  int lane = threadIdx.x & 31;
  int base = wave * 32;
  if (base >= E) return;                 // E % 32 == 0: full groups only
  int eIdx = base + lane;
  int r = (int)rows[eIdx];
  int c = (int)cols[eIdx];
  float nrm = dinv[r] * dinv[c];

#pragma unroll
  for (int i = 0; i < 32; ++i) {
    int ri = __shfl(r, i, 32);
    int ci = __shfl(c, i, 32);
    float ni = __shfl(nrm, i, 32);
    const float* hr = H + (size_t)ri * F;
    float* ac = agg + (size_t)ci * F;
#pragma unroll
    for (int f = lane; f < F; f += 32) atomicAdd(&ac[f], ni * hr[f]);
  }
}

// ---------------- epilogue: agg += dinv^2 * h + bias; optional ReLU ----------------
template <int F, bool RELU>
__global__ void k_finalize(const float* __restrict__ H, const float* __restrict__ dinv,
                           const float* __restrict__ bias, float* __restrict__ agg, int n) {
  int i = blockIdx.x * blockDim.x + threadIdx.x;
  int stride = gridDim.x * blockDim.x;
  int total = n * F;
  for (; i < total; i += stride) {
    int node = i / F;                    // compile-time F -> shift/magic mul
    int f = i - node * F;
    float di = dinv[node];
    float v = agg[i] + di * di * H[i] + bias[f];
    if (RELU) v = fmaxf(v, 0.0f);
    agg[i] = v;
  }
}

// ---------------- launch ----------------

static inline int cdiv(int a, int b) { return (a + b - 1) / b; }

template <int FIN, int FOUT, bool RELU>
static void run_layer(const float* in, const float* W, const float* bias,
                      float* hbuf, float* agg, const long long* rows,
                      const long long* cols, const float* dinv, hipStream_t stream) {
  constexpr int FOUT_P = (FOUT + 15) & ~15;
  int totalTiles = (NN / 16) * (FOUT_P / 16);
  k_gemm_wmma<FIN, FOUT><<<cdiv(totalTiles, 8), 256, 0, stream>>>(in, W, hbuf, totalTiles);
  int nel = NN * FOUT;
  k_zero<<<cdiv(nel, 256), 256, 0, stream>>>(agg, nel);
  k_scatter<FOUT><<<cdiv(NE / 32, 8), 256, 0, stream>>>(rows, cols, dinv, hbuf, agg, NE);
  k_finalize<FOUT, RELU><<<cdiv(nel, 256), 256, 0, stream>>>(hbuf, dinv, bias, agg, NN);
}

extern "C" void kernel_launch(void* const* d_in, const int* in_sizes, int n_in,
                              void* d_out, int out_size, void* d_ws, size_t ws_size,
                              hipStream_t stream) {
  (void)in_sizes; (void)n_in; (void)out_size; (void)ws_size;

  const float*     x    = (const float*)d_in[0];
  const long long* ei   = (const long long*)d_in[1];  // int64 [2, E]
  const long long* rows = ei;
  const long long* cols = ei + NE;
  const float* W1 = (const float*)d_in[2];
  const float* B1 = (const float*)d_in[3];
  const float* W2 = (const float*)d_in[4];
  const float* B2 = (const float*)d_in[5];
  const float* W3 = (const float*)d_in[6];
  const float* B3 = (const float*)d_in[7];
  const float* W4 = (const float*)d_in[8];
  const float* B4 = (const float*)d_in[9];

  // workspace: dinv (N f32) | bufH (N*128 f32) | bufG (N*128 f32)
  float* dinv = (float*)d_ws;
  size_t offH = ((size_t)NN * 4 + 255) & ~(size_t)255;
  float* bufH = (float*)((char*)d_ws + offH);
  float* bufG = bufH + (size_t)NN * 128;

  // degrees + dinv (bufG as scratch)
  k_zero<<<cdiv(NN, 256), 256, 0, stream>>>(bufG, NN);
  k_degree<<<cdiv(NE, 256), 256, 0, stream>>>(cols, bufG, NE);
  k_dinv<<<cdiv(NN, 256), 256, 0, stream>>>(bufG, dinv, NN);

  // L1: x(21) -> G(32); input buffer dead after GEMM, so agg may alias next input
  run_layer<21, 32, true>(x, W1, B1, bufH, bufG, rows, cols, dinv, stream);
  // L2: G(32) -> G(64)
  run_layer<32, 64, true>(bufG, W2, B2, bufH, bufG, rows, cols, dinv, stream);
  // L3: G(64) -> G(128)
  run_layer<64, 128, true>(bufG, W3, B3, bufH, bufG, rows, cols, dinv, stream);
  // L4: G(128) -> d_out(21), no ReLU
  run_layer<128, 21, false>(bufG, W4, B4, bufH, (float*)d_out, rows, cols, dinv, stream);
}